// Ffjord_43602507989007
// MI455X (gfx1250) — compile-verified
//
#include <hip/hip_runtime.h>

// FFJORD fused integrator for MI455X (gfx1250, wave32, WMMA f32 16x16x4).
// One kernel does all 5 RK4 steps x 4 func evals. Weights live in LDS in a
// fragment-major swizzled layout (each WMMA B fragment = one ds_load_b64 into
// an adjacent register pair); activations relayout D->A through per-wave LDS
// staging buffers (conflict-free strides).

typedef __attribute__((ext_vector_type(2))) float v2f;
typedef __attribute__((ext_vector_type(8))) float v8f;

#define NWAVES    8
#define MROWS     16
#define ROWS_BLK  (NWAVES * MROWS)   // 128 rows per workgroup
#define DD        32
#define HH        128
#define SSTRIDE   132                // stage stride (conflict-free for D-store and A-load patterns)

// Swizzled (fragment-major) weight blocks: size = K*N floats, fragment f =
// (kc*NT + nt) occupies 64 floats: [hi*32 + 2*lo + v]  (v = 0/1 = K parity).
#define OFF_W0   0                         // 8 kc * 8 nt * 64 = 4096
#define OFF_W1   (OFF_W0 + 4096)           // 32 kc * 8 nt * 64 = 16384
#define OFF_W2   (OFF_W1 + 16384)          // 32 kc * 2 nt * 64 = 4096
#define OFF_B0   (OFF_W2 + 4096)           // 24576
#define OFF_T0   (OFF_B0 + 128)
#define OFF_B1   (OFF_T0 + 128)
#define OFF_T1   (OFF_B1 + 128)
#define OFF_B2   (OFF_T1 + 128)
#define OFF_T2   (OFF_B2 + 32)
#define OFF_STG  (OFF_T2 + 32)             // 25152
#define STG_WAVE (2 * 16 * SSTRIDE)        // h-buffer + u-buffer per wave (4224 floats)
#define SMEM_FLOATS (OFF_STG + NWAVES * STG_WAVE)  // 58944 floats = 235776 B < 320 KB

__device__ __forceinline__ float fast_tanh(float x) {
  float cx = fminf(fmaxf(x, -10.0f), 10.0f);
  float e  = __expf(2.0f * cx);                    // v_exp_f32 (TRANS)
  return 1.0f - 2.0f * __builtin_amdgcn_rcpf(e + 1.0f);  // v_rcp_f32, no IEEE div
}

__device__ __forceinline__ v8f wmma4(v2f a, v2f b, v8f c) {
  // D(16x16,f32) += A(16x4,f32) * B(4x16,f32)
  return __builtin_amdgcn_wmma_f32_16x16x4_f32(false, a, false, b, (short)0, c,
                                               false, false);
}

// One evaluation of f(z,t) and its JVP + Hutchinson trace for 16 rows/wave.
__device__ __forceinline__ void func_eval(
    const float* smem, float* sH, float* sU, int lo, int hi,
    const v2f* zin, const v2f* eps_a, const float (*eps_d)[8],
    v2f* kcur, float* lsum, float t, float wrk)
{
  const v8f ZERO8 = {0.f,0.f,0.f,0.f,0.f,0.f,0.f,0.f};

  // Per-lane base pointers (DS immediate offsets do the rest).
  const int fb = hi * 32 + 2 * lo;                 // fragment-base lane offset
  const float* wF0 = smem + OFF_W0 + fb;
  const float* wF1 = smem + OFF_W1 + fb;
  const float* wF2 = smem + OFF_W2 + fb;
  float* sHa = sH + lo * SSTRIDE + 2 * hi;         // A-fragment load base
  float* sUa = sU + lo * SSTRIDE + 2 * hi;
  float* sHs = sH + 8 * hi * SSTRIDE + lo;         // D-layout store base
  float* sUs = sU + 8 * hi * SSTRIDE + lo;

  // ----- layer 0 primal: h1 = tanh(z @ W0a + (b0 + t*w0t)) -----
  v8f acc[8];
#pragma unroll
  for (int nt = 0; nt < 8; ++nt) acc[nt] = ZERO8;
  for (int kc = 0; kc < 8; ++kc) {
    v2f a = zin[kc];
#pragma unroll
    for (int nt = 0; nt < 8; ++nt) {
      v2f b = *(const v2f*)(wF0 + (kc * 8 + nt) * 64);   // ds_load_b64
      acc[nt] = wmma4(a, b, acc[nt]);
    }
  }
#pragma unroll
  for (int nt = 0; nt < 8; ++nt) {
    int col = 16 * nt + lo;
    float bias = smem[OFF_B0 + col] + t * smem[OFF_T0 + col];
#pragma unroll
    for (int v = 0; v < 8; ++v)
      sHs[v * SSTRIDE + 16 * nt] = fast_tanh(acc[nt][v] + bias);
  }
  __builtin_amdgcn_wave_barrier();

  // ----- layer 0 tangent: u1 = (eps @ W0a) * (1 - h1^2) -----
#pragma unroll
  for (int nt = 0; nt < 8; ++nt) acc[nt] = ZERO8;
  for (int kc = 0; kc < 8; ++kc) {
    v2f a = eps_a[kc];
#pragma unroll
    for (int nt = 0; nt < 8; ++nt) {
      v2f b = *(const v2f*)(wF0 + (kc * 8 + nt) * 64);
      acc[nt] = wmma4(a, b, acc[nt]);
    }
  }
#pragma unroll
  for (int nt = 0; nt < 8; ++nt) {
#pragma unroll
    for (int v = 0; v < 8; ++v) {
      int idx = v * SSTRIDE + 16 * nt;
      float h = sHs[idx];                          // same-lane reload
      sUs[idx] = acc[nt][v] * (1.0f - h * h);
    }
  }
  __builtin_amdgcn_wave_barrier();

  // ----- layer 1 primal: h2 = tanh(h1 @ W1a + (b1 + t*w1t)) -----
#pragma unroll
  for (int nt = 0; nt < 8; ++nt) acc[nt] = ZERO8;
  for (int kc = 0; kc < 32; ++kc) {
    v2f a = *(const v2f*)(sHa + 4 * kc);           // ds_load_b64, conflict free
#pragma unroll
    for (int nt = 0; nt < 8; ++nt) {
      v2f b = *(const v2f*)(wF1 + (kc * 8 + nt) * 64);
      acc[nt] = wmma4(a, b, acc[nt]);
    }
  }
  __builtin_amdgcn_wave_barrier();                 // all h1 reads done before overwrite
#pragma unroll
  for (int nt = 0; nt < 8; ++nt) {
    int col = 16 * nt + lo;
    float bias = smem[OFF_B1 + col] + t * smem[OFF_T1 + col];
#pragma unroll
    for (int v = 0; v < 8; ++v)
      sHs[v * SSTRIDE + 16 * nt] = fast_tanh(acc[nt][v] + bias);
  }
  __builtin_amdgcn_wave_barrier();

  // ----- layer 1 tangent: u2 = (u1 @ W1a) * (1 - h2^2) -----
#pragma unroll
  for (int nt = 0; nt < 8; ++nt) acc[nt] = ZERO8;
  for (int kc = 0; kc < 32; ++kc) {
    v2f a = *(const v2f*)(sUa + 4 * kc);
#pragma unroll
    for (int nt = 0; nt < 8; ++nt) {
      v2f b = *(const v2f*)(wF1 + (kc * 8 + nt) * 64);
      acc[nt] = wmma4(a, b, acc[nt]);
    }
  }
  __builtin_amdgcn_wave_barrier();                 // all u1 reads done before overwrite
#pragma unroll
  for (int nt = 0; nt < 8; ++nt) {
#pragma unroll
    for (int v = 0; v < 8; ++v) {
      int idx = v * SSTRIDE + 16 * nt;
      float h = sHs[idx];
      sUs[idx] = acc[nt][v] * (1.0f - h * h);
    }
  }
  __builtin_amdgcn_wave_barrier();

  // ----- layer 2: dz = h2 @ W2a + (b2 + t*w2t); Jeps = u2 @ W2a -----
  v8f dacc[2], jacc[2];
  dacc[0] = ZERO8; dacc[1] = ZERO8; jacc[0] = ZERO8; jacc[1] = ZERO8;
  for (int kc = 0; kc < 32; ++kc) {
    v2f ah = *(const v2f*)(sHa + 4 * kc);
    v2f au = *(const v2f*)(sUa + 4 * kc);
#pragma unroll
    for (int nt = 0; nt < 2; ++nt) {
      v2f b = *(const v2f*)(wF2 + (kc * 2 + nt) * 64);
      dacc[nt] = wmma4(ah, b, dacc[nt]);
      jacc[nt] = wmma4(au, b, jacc[nt]);           // B fragment reused for both paths
    }
  }
  __builtin_amdgcn_wave_barrier();

  // Hutchinson trace: sum over columns of eps .* Jeps, per row
  float p[8];
#pragma unroll
  for (int v = 0; v < 8; ++v)
    p[v] = eps_d[0][v] * jacc[0][v] + eps_d[1][v] * jacc[1][v];
#pragma unroll
  for (int off = 1; off < 16; off <<= 1) {
#pragma unroll
    for (int v = 0; v < 8; ++v) p[v] += __shfl_xor(p[v], off, 32);
  }
#pragma unroll
  for (int v = 0; v < 8; ++v) lsum[v] += wrk * p[v];

  // stage dz (D-layout) and reload as A-fragments -> kcur
#pragma unroll
  for (int nt = 0; nt < 2; ++nt) {
    int col = 16 * nt + lo;
    float bias = smem[OFF_B2 + col] + t * smem[OFF_T2 + col];
#pragma unroll
    for (int v = 0; v < 8; ++v)
      sHs[v * SSTRIDE + 16 * nt] = dacc[nt][v] + bias;
  }
  __builtin_amdgcn_wave_barrier();
#pragma unroll
  for (int kc = 0; kc < 8; ++kc)
    kcur[kc] = *(const v2f*)(sHa + 4 * kc);
  __builtin_amdgcn_wave_barrier();
}

__global__ __launch_bounds__(256)
void ffjord_kernel(const float* __restrict__ z1, const float* __restrict__ eps,
                   const float* __restrict__ W0, const float* __restrict__ b0,
                   const float* __restrict__ W1, const float* __restrict__ b1,
                   const float* __restrict__ W2, const float* __restrict__ b2,
                   float* __restrict__ out_z, float* __restrict__ out_ld)
{
  extern __shared__ float smem[];
  const int tid  = threadIdx.x;
  const int lane = tid & 31;
  const int wave = tid >> 5;
  const int lo   = lane & 15;
  const int hi   = lane >> 4;

  // ---- cooperative weight load + fragment-major swizzle ----
  // frag idx for (k,n): ((k>>2)*NT + (n>>4))*64 + ((k>>1)&1)*32 + 2*(n&15) + (k&1)
  for (int i = tid; i < 32 * 128; i += blockDim.x) {
    int k = i >> 7, n = i & 127;
    int idx = (((k >> 2) * 8 + (n >> 4)) << 6) + (((k >> 1) & 1) << 5) + 2 * (n & 15) + (k & 1);
    smem[OFF_W0 + idx] = W0[k * 128 + n];
  }
  for (int i = tid; i < 128 * 128; i += blockDim.x) {
    int k = i >> 7, n = i & 127;
    int idx = (((k >> 2) * 8 + (n >> 4)) << 6) + (((k >> 1) & 1) << 5) + 2 * (n & 15) + (k & 1);
    smem[OFF_W1 + idx] = W1[k * 128 + n];
  }
  for (int i = tid; i < 128 * 32; i += blockDim.x) {
    int k = i >> 5, n = i & 31;
    int idx = (((k >> 2) * 2 + (n >> 4)) << 6) + (((k >> 1) & 1) << 5) + 2 * (n & 15) + (k & 1);
    smem[OFF_W2 + idx] = W2[k * 32 + n];
  }
  for (int n = tid; n < 128; n += blockDim.x) {
    smem[OFF_T0 + n] = W0[32 * 128 + n];
    smem[OFF_B0 + n] = b0[n];
    smem[OFF_T1 + n] = W1[128 * 128 + n];
    smem[OFF_B1 + n] = b1[n];
  }
  for (int n = tid; n < 32; n += blockDim.x) {
    smem[OFF_T2 + n] = W2[128 * 32 + n];
    smem[OFF_B2 + n] = b2[n];
  }
  __syncthreads();

  float* sH = smem + OFF_STG + wave * STG_WAVE;
  float* sU = sH + 16 * SSTRIDE;
  const int rowbase = blockIdx.x * ROWS_BLK + wave * MROWS;

  // ---- resident state ----
  v2f z_a[8], eps_a[8], zin[8], ksum[8], kcur[8];
  float eps_d[2][8];
  float ldacc[8];

  {
    const float* zrow = z1  + (size_t)(rowbase + lo) * DD + 2 * hi;
    const float* erow = eps + (size_t)(rowbase + lo) * DD + 2 * hi;
#pragma unroll
    for (int kc = 0; kc < 8; ++kc) {
      z_a[kc]   = *(const v2f*)(zrow + 4 * kc);
      eps_a[kc] = *(const v2f*)(erow + 4 * kc);
    }
#pragma unroll
    for (int nt = 0; nt < 2; ++nt)
#pragma unroll
      for (int v = 0; v < 8; ++v)
        eps_d[nt][v] = eps[(size_t)(rowbase + v + 8 * hi) * DD + 16 * nt + lo];
#pragma unroll
    for (int v = 0; v < 8; ++v) ldacc[v] = 0.0f;
  }

  // ---- RK4 backward-time integration: t = 1.0 -> 0.0, hstep = -0.2 ----
  const float hstep = -0.2f;
  for (int s = 0; s < 5; ++s) {
    const float tb = 1.0f - 0.2f * (float)s;
    float lsum[8];
#pragma unroll
    for (int v = 0; v < 8; ++v) lsum[v] = 0.0f;

#pragma unroll
    for (int i = 0; i < 8; ++i) zin[i] = z_a[i];
    func_eval(smem, sH, sU, lo, hi, zin, eps_a, eps_d, kcur, lsum, tb, 1.0f);
#pragma unroll
    for (int i = 0; i < 8; ++i) { ksum[i] = kcur[i]; zin[i] = z_a[i] + kcur[i] * (0.5f * hstep); }

    func_eval(smem, sH, sU, lo, hi, zin, eps_a, eps_d, kcur, lsum, tb + 0.5f * hstep, 2.0f);
#pragma unroll
    for (int i = 0; i < 8; ++i) { ksum[i] += kcur[i] * 2.0f; zin[i] = z_a[i] + kcur[i] * (0.5f * hstep); }

    func_eval(smem, sH, sU, lo, hi, zin, eps_a, eps_d, kcur, lsum, tb + 0.5f * hstep, 2.0f);
#pragma unroll
    for (int i = 0; i < 8; ++i) { ksum[i] += kcur[i] * 2.0f; zin[i] = z_a[i] + kcur[i] * hstep; }

    func_eval(smem, sH, sU, lo, hi, zin, eps_a, eps_d, kcur, lsum, tb + hstep, 1.0f);
#pragma unroll
    for (int i = 0; i < 8; ++i) ksum[i] += kcur[i];

    const float c = hstep / 6.0f;
#pragma unroll
    for (int i = 0; i < 8; ++i) z_a[i] += ksum[i] * c;   // z += dk
#pragma unroll
    for (int v = 0; v < 8; ++v) ldacc[v] -= c * lsum[v]; // ld -= dl
  }

  // ---- write outputs ----
  {
    float* orow = out_z + (size_t)(rowbase + lo) * DD + 2 * hi;
#pragma unroll
    for (int kc = 0; kc < 8; ++kc)
      *(v2f*)(orow + 4 * kc) = z_a[kc];
#pragma unroll
    for (int v = 0; v < 8; ++v)
      if (lo == v) out_ld[rowbase + 8 * hi + v] = ldacc[v];
  }
}

extern "C" void kernel_launch(void* const* d_in, const int* in_sizes, int n_in,
                              void* d_out, int out_size, void* d_ws, size_t ws_size,
                              hipStream_t stream) {
  (void)n_in; (void)out_size; (void)d_ws; (void)ws_size;
  const float* z1  = (const float*)d_in[0];
  const float* eps = (const float*)d_in[1];
  const float* W0  = (const float*)d_in[2];
  const float* b0  = (const float*)d_in[3];
  const float* W1  = (const float*)d_in[4];
  const float* b1  = (const float*)d_in[5];
  const float* W2  = (const float*)d_in[6];
  const float* b2  = (const float*)d_in[7];

  const int Btot = in_sizes[0] / DD;           // 131072
  float* out_z  = (float*)d_out;
  float* out_ld = out_z + (size_t)Btot * DD;

  dim3 grid(Btot / ROWS_BLK), block(NWAVES * 32);
  size_t shmem = (size_t)SMEM_FLOATS * sizeof(float);   // ~236 KB of the 320 KB WGP LDS
  ffjord_kernel<<<grid, block, shmem, stream>>>(z1, eps, W0, b0, W1, b1, W2, b2,
                                                out_z, out_ld);
}